// AfmoeTokenChoiceRouter_8186207666192
// MI455X (gfx1250) — compile-verified
//
#include <hip/hip_runtime.h>
#include <cstddef>
#include <cstdint>

typedef __attribute__((ext_vector_type(2))) float v2f;
typedef __attribute__((ext_vector_type(8))) float v8f;
typedef int b128_t __attribute__((vector_size(16)));          // matches builtin param
typedef __attribute__((address_space(1))) b128_t* gptr128_t;  // global int4*
typedef __attribute__((address_space(3))) b128_t* lptr128_t;  // LDS int4*

constexpr int kH       = 2048;  // hidden dim
constexpr int kE       = 64;    // experts
constexpr int kTopK    = 8;
constexpr int kCK      = 32;    // K chunk staged in LDS
constexpr int kNChunks = kH / kCK;              // 64
constexpr int kTileT   = 64;    // tokens per workgroup
constexpr int kStrideF = 36;    // padded LDS row stride (floats): 36*l mod 64 distinct
constexpr int kTileF   = kTileT * kStrideF;     // 2304 floats per matrix tile
constexpr int kBufF    = 2 * kTileF;            // A+B tile per buffer
constexpr int kSStr    = 65;    // score row stride (floats)

#if defined(__AMDGCN__) && __has_builtin(__builtin_amdgcn_global_load_async_to_lds_b128) && __has_builtin(__builtin_amdgcn_s_wait_asynccnt)
#define HAS_ASYNC_LDS 1
#else
#define HAS_ASYNC_LDS 0
#endif

__device__ __forceinline__ void copy_b128(const float* g, float* l) {
#if HAS_ASYNC_LDS
    // global -> LDS DMA, tracked by ASYNCcnt; no VGPR round-trip
    __builtin_amdgcn_global_load_async_to_lds_b128(
        (gptr128_t)(uintptr_t)g,
        (lptr128_t)(uint32_t)(uintptr_t)l,
        0, 0);
#else
    *(float4*)l = *(const float4*)g;
#endif
}

template <int N>
__device__ __forceinline__ void wait_async() {
#if HAS_ASYNC_LDS
    __builtin_amdgcn_s_wait_asynccnt(N);
#endif
}

__global__ __launch_bounds__(128) void afmoe_router_kernel(
    const float* __restrict__ x, const float* __restrict__ bias,
    const float* __restrict__ w, float* __restrict__ outS,
    int* __restrict__ outI)
{
    __shared__ float smem[2 * kBufF];   // 36,864 bytes (double-buffered A+B tiles)

    const int tid  = threadIdx.x;
    const int wave = tid >> 5;     // 0..3 -> token sub-tile
    const int lane = tid & 31;
    const int half = lane >> 4;    // 0: K,K+1   1: K+2,K+3
    const int lm   = lane & 15;    // M (token) / N (expert) index within tile
    const int tok0 = blockIdx.x * kTileT;

    // per-lane expert bias for the 4 N-tiles (expert = nt*16 + lm)
    float biasv[4];
    #pragma unroll
    for (int nt = 0; nt < 4; ++nt) biasv[nt] = bias[nt * 16 + lm];

    v8f acc[4];
    #pragma unroll
    for (int nt = 0; nt < 4; ++nt)
        acc[nt] = (v8f){0.f, 0.f, 0.f, 0.f, 0.f, 0.f, 0.f, 0.f};

    // staging map: 64 rows x 32 floats per matrix; 8 threads per row, 4 passes
    const int lrow = tid >> 3;        // 0..15 (16 rows per pass)
    const int lcol = (tid & 7) * 4;   // 0..28

    // issue one chunk's async loads (8 b128 per thread => ASYNCcnt += 8 per wave)
    auto stage = [&](int kc) {
        float* bufA = smem + (kc & 1) * kBufF;
        float* bufB = bufA + kTileF;
        const int kbase = kc * kCK;
        #pragma unroll
        for (int p = 0; p < 4; ++p) {
            const int r = p * 16 + lrow;
            copy_b128(x + (size_t)(tok0 + r) * kH + kbase + lcol,
                      &bufA[r * kStrideF + lcol]);
            copy_b128(w + (size_t)r * kH + kbase + lcol,
                      &bufB[r * kStrideF + lcol]);
        }
    };

    stage(0);   // prologue

    for (int kc = 0; kc < kNChunks; ++kc) {
        if (kc + 1 < kNChunks) {
            stage(kc + 1);        // overlap next chunk's DMA with this chunk's WMMA
            wait_async<8>();      // only next chunk's 8 loads may remain in flight
        } else {
            wait_async<0>();
        }
        __syncthreads();          // all waves' tile data resident in LDS

        const float* curA = smem + (kc & 1) * kBufF;
        const float* curB = curA + kTileF;
        const float* aRow = &curA[(wave * 16 + lm) * kStrideF + 2 * half];

        #pragma unroll
        for (int kk = 0; kk < kCK; kk += 4) {
            // A fragment: 16x4 f32, lanes 0-15 hold K=kk,kk+1; lanes 16-31 K=kk+2,kk+3
            const v2f a = *(const v2f*)&aRow[kk];
            #pragma unroll
            for (int nt = 0; nt < 4; ++nt) {
                // B fragment: 4x16 f32, B[k][n] = gate_w[n][k]
                const v2f b = *(const v2f*)&curB[(nt * 16 + lm) * kStrideF + kk + 2 * half];
                acc[nt] = __builtin_amdgcn_wmma_f32_16x16x4_f32(
                    false, a, false, b, (short)0, acc[nt], false, false);
            }
        }
        __syncthreads();          // done reading this buffer before it is re-staged
    }

    // ---- phase 2: sigmoid + biased top-k (reuse LDS) ----
    float* ldsS = smem;                    // raw sigmoid scores  [64][65]
    float* ldsQ = smem + kTileT * kSStr;   // biased scores       [64][65]

    // C layout: VGPR r -> M = r + 8*half, lane%16 -> N
    #pragma unroll
    for (int nt = 0; nt < 4; ++nt) {
        #pragma unroll
        for (int r = 0; r < 8; ++r) {
            const int tl = wave * 16 + r + 8 * half;   // local token
            const int e  = nt * 16 + lm;               // expert
            // sigmoid via v_exp + v_rcp (no IEEE-division expansion)
            const float s = __builtin_amdgcn_rcpf(1.0f + __expf(-acc[nt][r]));
            ldsS[tl * kSStr + e] = s;
            ldsQ[tl * kSStr + e] = s + biasv[nt];
        }
    }
    __syncthreads();

    if (tid < kTileT) {
        float* rowS = &ldsS[tid * kSStr];
        float* rowQ = &ldsQ[tid * kSStr];
        float sc[kTopK]; int id[kTopK];
        float sum = 0.f;
        #pragma unroll
        for (int j = 0; j < kTopK; ++j) {
            float best = -1e30f; int bi = 0;
            for (int e = 0; e < kE; ++e) {
                const float v = rowQ[e];
                if (v > best) { best = v; bi = e; }   // strict > == lowest-index tie-break
            }
            const float s = rowS[bi];
            rowQ[bi] = -1e30f;                        // exclude for next pass
            sc[j] = s; id[j] = bi; sum += s;
        }
        const float scl = 2.5f * __builtin_amdgcn_rcpf(sum + 1e-20f);
        const size_t tok = (size_t)tok0 + tid;
        #pragma unroll
        for (int j = 0; j < kTopK; ++j) {
            outS[tok * kTopK + j] = sc[j] * scl;
            outI[tok * kTopK + j] = id[j];
        }
    }
}

extern "C" void kernel_launch(void* const* d_in, const int* in_sizes, int n_in,
                              void* d_out, int out_size, void* d_ws, size_t ws_size,
                              hipStream_t stream) {
    (void)n_in; (void)out_size; (void)d_ws; (void)ws_size;
    const float* x    = (const float*)d_in[0];   // hidden_states [T, 2048]
    const float* bias = (const float*)d_in[1];   // expert_bias   [64]
    const float* w    = (const float*)d_in[2];   // gate_w        [64, 2048]
    const int T = in_sizes[0] / kH;              // 16384

    float* outS = (float*)d_out;                       // top_scores [T, 8] f32
    int*   outI = (int*)d_out + (size_t)T * kTopK;     // selected   [T, 8] i32

    const int blocks = T / kTileT;                     // 256
    afmoe_router_kernel<<<blocks, 128, 0, stream>>>(x, bias, w, outS, outI);
}